// CAM_Module_50483045597337
// MI455X (gfx1250) — compile-verified
//
#include <hip/hip_runtime.h>
#include <hip/hip_bf16.h>

// CAM module: B=16, C=64, N=H*W=65536, fp32.
// energy = Q Q^T (per batch), att = softmax(rowmax - energy) = exp(rowmin - e)/sum,
// out = gamma * (att Q) + x.

#define B_DIM 16
#define C_DIM 64
#define N_DIM 65536
#define LDS_STRIDE 68   // 64 + 4 pad: keeps 16B rows aligned for B128 async stores
                        // and bank = (4*row + col) % 64 conflict-free for frags

typedef float v2f __attribute__((ext_vector_type(2)));
typedef float v8f __attribute__((ext_vector_type(8)));
typedef int   v4i __attribute__((ext_vector_type(4)));

#define GLB_AS __attribute__((address_space(1)))
#define LDS_AS __attribute__((address_space(3)))

#if __has_builtin(__builtin_amdgcn_global_load_async_to_lds_b128)
#define HAS_ASYNC_LDS 1
#else
#define HAS_ASYNC_LDS 0
#endif

#if HAS_ASYNC_LDS
#if __has_builtin(__builtin_amdgcn_s_wait_asynccnt)
#define WAIT_ASYNC(n) __builtin_amdgcn_s_wait_asynccnt(n)
#else
#define WAIT_ASYNC(n) asm volatile("s_wait_asynccnt %0" ::"i"(n) : "memory")
#endif
#else
#define WAIT_ASYNC(n) ((void)0)
#endif

// Copy 16 bytes global -> LDS. Async (no VGPR staging, ASYNCcnt-tracked) when
// the gfx1250 builtin exists, else load+store fallback (proven codegen).
__device__ __forceinline__ void copy16_g2l(const float* __restrict__ g, float* l) {
#if HAS_ASYNC_LDS
    __builtin_amdgcn_global_load_async_to_lds_b128(
        (GLB_AS v4i*)g, (LDS_AS v4i*)l, 0, 0);
#else
    const float4 v = *(const float4*)g;
    l[0] = v.x; l[1] = v.y; l[2] = v.z; l[3] = v.w;
#endif
}

__global__ void cam_zero_kernel(float* __restrict__ p, int n) {
    int i = blockIdx.x * blockDim.x + threadIdx.x;
    if (i < n) p[i] = 0.0f;
}

// ---------------------------------------------------------------------------
// Kernel 1: partial Gram matrices. grid = (32 n-chunks, 16 batches), 512 thr.
// Double-buffered async tile loads; each wave owns one 16x16 tile (ci,cj) of
// the 64x64 energy matrix, chained v_wmma_f32_16x16x4_f32 over the pixel dim.
// ---------------------------------------------------------------------------
__global__ __launch_bounds__(512) void cam_gram_kernel(const float* __restrict__ x,
                                                       float* __restrict__ energy) {
    __shared__ float ldsX[2][C_DIM * LDS_STRIDE];

    const int b    = blockIdx.y;
    const int t    = threadIdx.x;
    const int lane = t & 31;
    const int wave = t >> 5;            // 0..15
    const int ci   = wave >> 2;         // row tile 0..3
    const int cj   = wave & 3;          // col tile 0..3
    const int half = lane >> 4;         // 0/1
    const int l16  = lane & 15;
    const int kb   = half * 2;          // K sub-offset per ISA A/B layout

    const int tn   = t & 15;            // float4 group within 64 cols
    const int trow = t >> 4;            // 0..31

    const float* xb = x + (size_t)b * C_DIM * N_DIM;

    auto issue_tile = [&](int n0, int buf) {
#pragma unroll
        for (int h = 0; h < 2; ++h) {
            const int row = trow + h * 32;
            copy16_g2l(xb + (size_t)row * N_DIM + n0 + tn * 4,
                       &ldsX[buf][row * LDS_STRIDE + tn * 4]);
        }
    };

    v8f acc = {0.f, 0.f, 0.f, 0.f, 0.f, 0.f, 0.f, 0.f};

    const int n_start = blockIdx.x * 2048;
    const int n_end   = n_start + 2048;

    issue_tile(n_start, 0);
    int buf = 0;
    for (int n0 = n_start; n0 < n_end; n0 += 64) {
        if (n0 + 64 < n_end) {
            issue_tile(n0 + 64, buf ^ 1);   // prefetch next tile into other buffer
            WAIT_ASYNC(2);                  // wait only for the current tile's 2 loads
        } else {
            WAIT_ASYNC(0);
        }
        __syncthreads();

        const float* arow = &ldsX[buf][(ci * 16 + l16) * LDS_STRIDE + kb];
        const float* brow = &ldsX[buf][(cj * 16 + l16) * LDS_STRIDE + kb];
#pragma unroll
        for (int k = 0; k < 64; k += 4) {
            v2f a  = {arow[k], arow[k + 1]};
            v2f bb = {brow[k], brow[k + 1]};
            acc = __builtin_amdgcn_wmma_f32_16x16x4_f32(
                false, a, false, bb, (short)0, acc, false, false);
        }
        __syncthreads();
        buf ^= 1;
    }

    // accumulate this block's partial tile into the global energy matrix
    const int col = cj * 16 + l16;
    float* edst = energy + ((size_t)b * C_DIM + ci * 16 + half * 8) * C_DIM + col;
#pragma unroll
    for (int r = 0; r < 8; ++r) {
        unsafeAtomicAdd(edst + (size_t)r * C_DIM, acc[r]);
    }
}

// ---------------------------------------------------------------------------
// Kernel 2: attention = softmax(rowmax - energy) == exp(rowmin - e) / sum.
// One wave32 per (b,c) row of 64; shfl_xor reductions.
// ---------------------------------------------------------------------------
__global__ __launch_bounds__(256) void cam_softmax_kernel(const float* __restrict__ energy,
                                                          float* __restrict__ att) {
    const int gid  = blockIdx.x * 8 + (threadIdx.x >> 5);  // row id 0..1023
    const int lane = threadIdx.x & 31;
    const float* row = energy + (size_t)gid * C_DIM;

    float e0 = row[lane];
    float e1 = row[lane + 32];

    float m = fminf(e0, e1);
#pragma unroll
    for (int off = 16; off > 0; off >>= 1) m = fminf(m, __shfl_xor(m, off, 32));

    float v0 = __expf(m - e0);
    float v1 = __expf(m - e1);
    float s = v0 + v1;
#pragma unroll
    for (int off = 16; off > 0; off >>= 1) s += __shfl_xor(s, off, 32);

    const float inv = 1.0f / s;
    att[(size_t)gid * C_DIM + lane]      = v0 * inv;
    att[(size_t)gid * C_DIM + lane + 32] = v1 * inv;
}

// ---------------------------------------------------------------------------
// Kernel 3: out[b,c,n] = gamma * sum_d att[b,c,d] * x[b,d,n] + x[b,c,n].
// grid = (1024 n-tiles, 16 batches), 512 thr. Each wave: one 16(c) x 16(n)
// output tile, K=64 channel reduction via 16 chained WMMAs. Both the att
// matrix and the x tile arrive via async-to-LDS.
// ---------------------------------------------------------------------------
__global__ __launch_bounds__(512) void cam_out_kernel(const float* __restrict__ x,
                                                      const float* __restrict__ att,
                                                      const float* __restrict__ gamma_p,
                                                      float* __restrict__ out) {
    __shared__ float ldsA[C_DIM * LDS_STRIDE];  // attention [c][d]
    __shared__ float ldsX[C_DIM * LDS_STRIDE];  // x tile    [d][n_local]

    const int b    = blockIdx.y;
    const int n0   = blockIdx.x * 64;
    const int t    = threadIdx.x;
    const int lane = t & 31;
    const int wave = t >> 5;
    const int ci   = wave >> 2;         // c tile 0..3
    const int sub  = wave & 3;          // n subtile 0..3
    const int half = lane >> 4;
    const int l16  = lane & 15;
    const int kb   = half * 2;

    const float gamma = gamma_p[0];

    {
        const int tn = t & 15, trow = t >> 4;
#pragma unroll
        for (int h = 0; h < 2; ++h) {
            const int row = trow + h * 32;
            copy16_g2l(x + ((size_t)b * C_DIM + row) * N_DIM + n0 + tn * 4,
                       &ldsX[row * LDS_STRIDE + tn * 4]);
            copy16_g2l(att + ((size_t)b * C_DIM + row) * C_DIM + tn * 4,
                       &ldsA[row * LDS_STRIDE + tn * 4]);
        }
    }
    WAIT_ASYNC(0);
    __syncthreads();

    v8f acc = {0.f, 0.f, 0.f, 0.f, 0.f, 0.f, 0.f, 0.f};
    const float* arow = &ldsA[(ci * 16 + l16) * LDS_STRIDE + kb];
    const int ncol = sub * 16 + l16;
#pragma unroll
    for (int k = 0; k < 64; k += 4) {
        v2f a  = {arow[k], arow[k + 1]};
        v2f bb = {ldsX[(k + kb) * LDS_STRIDE + ncol],
                  ldsX[(k + kb + 1) * LDS_STRIDE + ncol]};
        acc = __builtin_amdgcn_wmma_f32_16x16x4_f32(
            false, a, false, bb, (short)0, acc, false, false);
    }

    // epilogue: residual + gamma scale; x[b,c,n] re-read from LDS tile
#pragma unroll
    for (int r = 0; r < 8; ++r) {
        const int c = ci * 16 + half * 8 + r;
        const float val = gamma * acc[r] + ldsX[c * LDS_STRIDE + ncol];
        out[((size_t)b * C_DIM + c) * N_DIM + n0 + ncol] = val;
    }
}

extern "C" void kernel_launch(void* const* d_in, const int* in_sizes, int n_in,
                              void* d_out, int out_size, void* d_ws, size_t ws_size,
                              hipStream_t stream) {
    const float* x     = (const float*)d_in[0];
    const float* gamma = (const float*)d_in[1];
    float* out = (float*)d_out;

    float* energy = (float*)d_ws;                       // B*C*C = 65536 floats
    float* att    = energy + (size_t)B_DIM * C_DIM * C_DIM;

    const int e_elems = B_DIM * C_DIM * C_DIM;          // 65536
    cam_zero_kernel<<<(e_elems + 255) / 256, 256, 0, stream>>>(energy, e_elems);

    cam_gram_kernel<<<dim3(32, B_DIM), 512, 0, stream>>>(x, energy);

    cam_softmax_kernel<<<(B_DIM * C_DIM) / 8, 256, 0, stream>>>(energy, att);

    cam_out_kernel<<<dim3(N_DIM / 64, B_DIM), 512, 0, stream>>>(x, att, gamma, out);
}